// SE3Convolution_23596550324728
// MI455X (gfx1250) — compile-verified
//
#include <hip/hip_runtime.h>
#include <hip/hip_bf16.h>

typedef __attribute__((ext_vector_type(2))) float v2f;
typedef __attribute__((ext_vector_type(4))) float v4f;
typedef __attribute__((ext_vector_type(8))) float v8f;

// ---------------------------------------------------------------------------
// Problem constants (from the reference)
// ---------------------------------------------------------------------------
#define NB      8        // batch
#define NCH     72       // n_in == n_out channels
#define NCH_PAD 80       // padded to 5 WMMA tiles of 16
#define DIN     32       // input spatial
#define DOUT    28       // output spatial (VALID, 5^3 kernel)
#define TAPS    125      // 5*5*5
#define SPATIAL_TOTAL (NB * DOUT * DOUT * DOUT)   // 175616, divisible by 32
#define X_CH_STRIDE   (DIN * DIN * DIN)           // 32768 elements
#define OUT_CH_STRIDE (DOUT * DOUT * DOUT)        // 21952 elements
#define KSLICE (NCH_PAD * NCH)                    // 5760 floats per tap slice

// ---------------------------------------------------------------------------
// Stage 1: combine weight + 9 basis blocks into dense kernel, laid out as
//   kws[tap][co][ci]  with co padded to 80 (pad region zero-filled).
// ---------------------------------------------------------------------------
__global__ __launch_bounds__(256) void se3_combine_kernel(
    const float* __restrict__ w,
    const float* b00, const float* b01, const float* b02,
    const float* b10, const float* b11, const float* b12,
    const float* b20, const float* b21, const float* b22,
    float* __restrict__ kws)
{
    const int idx = blockIdx.x * 256 + threadIdx.x;
    const int total = TAPS * KSLICE;
    if (idx >= total) return;

    const int tap = idx / KSLICE;
    const int r   = idx % KSLICE;
    const int co  = r / NCH;
    const int ci  = r % NCH;

    if (co >= NCH) { kws[idx] = 0.0f; return; }   // zero the co padding

    int i, o, a, dO;
    if (co < 8)       { i = 0; o = co;            a = 0;            dO = 1; }
    else if (co < 32) { i = 1; int t = co - 8;  o = t / 3; a = t % 3; dO = 3; }
    else              { i = 2; int t = co - 32; o = t / 5; a = t % 5; dO = 5; }

    int j, ii, bb, dI;
    if (ci < 8)       { j = 0; ii = ci;            bb = 0;            dI = 1; }
    else if (ci < 32) { j = 1; int t = ci - 8;  ii = t / 3; bb = t % 3; dI = 3; }
    else              { j = 2; int t = ci - 32; ii = t / 5; bb = t % 5; dI = 5; }

    const float* bt[9] = {b00, b01, b02, b10, b11, b12, b20, b21, b22};
    const float* bp = bt[i * 3 + j];

    const int woff    = (i * 3 + j) * 256 + (o * 8 + ii) * 4;
    const int bbase   = (a * dI + bb) * TAPS + tap;
    const int bstride = dO * dI * TAPS;

    float acc = 0.0f;
#pragma unroll
    for (int k = 0; k < 4; ++k)
        acc += w[woff + k] * bp[bbase + k * bstride];

    kws[idx] = acc;
}

// ---------------------------------------------------------------------------
// Stage 2: implicit-GEMM 3D conv using V_WMMA_F32_16X16X4_F32.
//   D[co, s] = sum_{tap, ci} kws[tap][co][ci] * x[n(s), ci, z+kz, y+ky, x+kx]
//
// Block = 8 waves. All waves share the full co range 0..79, so the per-tap
// kernel slice (80x72 f32 = 23 KB) is staged once per block into LDS and
// A-fragments come from ds_load_b64 (8x less L2 traffic for kernel data).
// Each wave: 5 co tiles x 2 spatial tiles -> 10 accumulators, 10 WMMA/K-step,
// 2250 K-steps (125 taps x 18 ci/4).
// Fragment layouts (ISA 7.12.2, 32-bit):
//   A 16x4 : lane m=lane&15 -> row M; K = 2*(lane>>4) + vgpr
//   B 4x16 : lane m -> col N;          K = 2*(lane>>4) + vgpr
//   D 16x16: lane m -> col N;          row M = vgpr + 8*(lane>>4)
// ---------------------------------------------------------------------------
__global__ __launch_bounds__(256, 1) void se3_conv_wmma_kernel(
    const float* __restrict__ x,
    const float* __restrict__ kws,
    float* __restrict__ out)
{
    __shared__ float klds[KSLICE];        // 23040 B, one tap slice

    const int lane  = threadIdx.x & 31;
    const int m     = lane & 15;          // M-row for A / N-col for B and D
    const int kpair = lane >> 4;          // selects K {0,1} vs {2,3}
    const int wave  = blockIdx.x * (blockDim.x >> 5) + (threadIdx.x >> 5);

    const int sbase = wave * 32;          // 32 spatial positions per wave (<2^31)

    // Per-lane x base pointers for the 2 spatial tiles (channel 0, tap 0).
    const float* xb[2];
#pragma unroll
    for (int st = 0; st < 2; ++st) {
        const int s  = sbase + st * 16 + m;
        const int n  = s / OUT_CH_STRIDE;
        const int rs = s % OUT_CH_STRIDE;
        const int oz = rs / (DOUT * DOUT);
        const int r2 = rs % (DOUT * DOUT);
        const int oy = r2 / DOUT;
        const int ox = r2 % DOUT;
        xb[st] = x + (long)n * NCH * X_CH_STRIDE
                   + oz * (DIN * DIN) + oy * DIN + ox;
    }

    v8f acc[5][2];
#pragma unroll
    for (int ct = 0; ct < 5; ++ct)
#pragma unroll
        for (int st = 0; st < 2; ++st)
            acc[ct][st] = (v8f){0.f, 0.f, 0.f, 0.f, 0.f, 0.f, 0.f, 0.f};

    // Per-lane LDS read base for A fragments: row co = ct*16 + m, K-pair start.
    const int klbase = m * NCH + 2 * kpair;

#pragma unroll 1
    for (int tap = 0; tap < TAPS; ++tap) {
        // ---- cooperative stage of this tap's kernel slice into LDS --------
        {
            const v4f* ksrc = (const v4f*)(kws + tap * KSLICE);
            v4f* kdst = (v4f*)klds;
            for (int t = threadIdx.x; t < KSLICE / 4; t += 256)   // 1440 v4f
                kdst[t] = ksrc[t];
        }
        __syncthreads();

        const int kz = tap / 25;
        const int kr = tap % 25;
        const int ky = kr / 5;
        const int kx = kr % 5;
        const int xoff = kz * (DIN * DIN) + ky * DIN + kx;

        const float* xb0 = xb[0] + xoff;
        const float* xb1 = xb[1] + xoff;

#pragma unroll 3
        for (int c = 0; c < NCH; c += 4) {
            v2f A[5];
#pragma unroll
            for (int ct = 0; ct < 5; ++ct)
                A[ct] = *(const v2f*)(&klds[klbase + ct * (16 * NCH) + c]); // ds_load_b64

            const int cc = (c + 2 * kpair) * X_CH_STRIDE;
            v2f B0, B1;
            B0.x = xb0[cc];
            B0.y = xb0[cc + X_CH_STRIDE];
            B1.x = xb1[cc];
            B1.y = xb1[cc + X_CH_STRIDE];

#pragma unroll
            for (int ct = 0; ct < 5; ++ct) {
                acc[ct][0] = __builtin_amdgcn_wmma_f32_16x16x4_f32(
                    false, A[ct], false, B0, (short)0, acc[ct][0], false, false);
                acc[ct][1] = __builtin_amdgcn_wmma_f32_16x16x4_f32(
                    false, A[ct], false, B1, (short)0, acc[ct][1], false, false);
            }
        }
        __syncthreads();   // protect LDS before next tap's staging
    }

    // Store: lane holds col N = m (spatial), row M = v + 8*kpair (co).
#pragma unroll
    for (int st = 0; st < 2; ++st) {
        const int s  = sbase + st * 16 + m;
        const int n  = s / OUT_CH_STRIDE;
        const int rs = s % OUT_CH_STRIDE;
        float* ob = out + (long)n * NCH * OUT_CH_STRIDE + rs;
#pragma unroll
        for (int ct = 0; ct < 5; ++ct) {
#pragma unroll
            for (int v = 0; v < 8; ++v) {
                const int co = ct * 16 + v + 8 * kpair;
                if (co < NCH)
                    ob[co * OUT_CH_STRIDE] = acc[ct][st][v];
            }
        }
    }
}

// ---------------------------------------------------------------------------
// Launch
// ---------------------------------------------------------------------------
extern "C" void kernel_launch(void* const* d_in, const int* in_sizes, int n_in,
                              void* d_out, int out_size, void* d_ws, size_t ws_size,
                              hipStream_t stream) {
    const float* x   = (const float*)d_in[0];
    const float* w   = (const float*)d_in[1];
    const float* b00 = (const float*)d_in[2];
    const float* b01 = (const float*)d_in[3];
    const float* b02 = (const float*)d_in[4];
    const float* b10 = (const float*)d_in[5];
    const float* b11 = (const float*)d_in[6];
    const float* b12 = (const float*)d_in[7];
    const float* b20 = (const float*)d_in[8];
    const float* b21 = (const float*)d_in[9];
    const float* b22 = (const float*)d_in[10];

    float* kws = (float*)d_ws;            // 125*80*72 floats = 2.88 MB
    float* out = (float*)d_out;

    // Stage 1: build combined kernel in workspace.
    {
        const int total  = TAPS * KSLICE;          // 720,000
        const int blocks = (total + 255) / 256;
        se3_combine_kernel<<<blocks, 256, 0, stream>>>(
            w, b00, b01, b02, b10, b11, b12, b20, b21, b22, kws);
    }

    // Stage 2: WMMA implicit-GEMM conv. 175616/32 = 5488 waves, 8 waves/block.
    {
        const int waves  = SPATIAL_TOTAL / 32;     // 5488
        const int blocks = waves / 8;              // 686
        se3_conv_wmma_kernel<<<blocks, 256, 0, stream>>>(x, kws, out);
    }
}